// AbstractRNN_84155589197895
// MI455X (gfx1250) — compile-verified
//
#include <hip/hip_runtime.h>
#include <math.h>

// ---------------------------------------------------------------------------
// Vanilla RNN:  h_t = tanh(x_t @ W + h_{t-1} @ U + b),  y = all h_t
// B=32, T=2048, D=256, H=256, fp32.
//
// Phase 1 (parallel):  XW[b,t,:] = x[b,t,:] @ W + b   -> staged in d_out
// Phase 2 (serial-T):  2 persistent WGs, 16 batch rows each.
//   - U fragments register-resident (256 VGPRs/wave; occupancy irrelevant:
//     only 2 waves/SIMD alive, 1024-VGPR budget on wave32).
//   - h in LDS (stride-padded, conflict-free).
//   - XW double-buffered in registers: loads for step t+1 issued at the top
//     of step t, hiding the serial-chain memory latency behind the K loop.
//   - nonlinearity via CDNA5 hardware V_TANH_F32 (16 independent trans ops,
//     satisfying the 1-independent-op TRANS hazard rule before first use).
// Matrix math via V_WMMA_F32_16X16X4_F32 (fp32 A/B, fp32 accum, wave32).
// ---------------------------------------------------------------------------

typedef float v2f __attribute__((ext_vector_type(2)));
typedef float v8f __attribute__((ext_vector_type(8)));

#define B_ 32
#define T_ 2048
#define D_ 256
#define H_ 256

__device__ __forceinline__ float hw_tanhf(float x) {
#if __has_builtin(__builtin_amdgcn_tanhf)
    return __builtin_amdgcn_tanhf(x);
#else
    float y;
    asm("v_tanh_f32 %0, %1" : "=v"(y) : "v"(x));
    return y;
#endif
}

// ---------------------------------------------------------------------------
// Kernel 1: XW = X @ W + bias    (X: [B*T, D] row-major, W: [D, H] row-major)
// One wave computes a 16x64 output tile (4 N-tiles, reusing the A fragment).
// grid: 4096 M-tiles * 4 N-groups = 16384 waves = 2048 blocks of 8 waves.
// ---------------------------------------------------------------------------
__global__ __launch_bounds__(256) void rnn_xw_gemm(const float* __restrict__ X,
                                                   const float* __restrict__ W,
                                                   const float* __restrict__ bias,
                                                   float* __restrict__ XW) {
    const int lane = threadIdx.x & 31;
    const int wave = threadIdx.x >> 5;
    const int gw   = blockIdx.x * 8 + wave;   // global wave id
    const int m0   = (gw >> 2) * 16;          // 16-row tile of [B*T]
    const int ngrp = gw & 3;                  // 64-column group of H
    const int r    = lane & 15;
    const int half = lane >> 4;

    v8f acc[4] = {};                          // 4 N-tiles of 16x16 fp32 accum

    const float* xrow = X + (size_t)(m0 + r) * D_ + 2 * half;
    #pragma unroll 4
    for (int k0 = 0; k0 < D_; k0 += 4) {
        // A fragment: lane(half,r) holds X[m0+r][k0+2*half .. +1]
        v2f a;
        a.x = xrow[k0];
        a.y = xrow[k0 + 1];
        #pragma unroll
        for (int tn = 0; tn < 4; ++tn) {
            const int n0 = ngrp * 64 + tn * 16;
            // B fragment: lane(half,r) holds W[k0+2*half .. +1][n0+r]
            v2f bm;
            bm.x = W[(size_t)(k0 + 2 * half)     * H_ + n0 + r];
            bm.y = W[(size_t)(k0 + 2 * half + 1) * H_ + n0 + r];
            acc[tn] = __builtin_amdgcn_wmma_f32_16x16x4_f32(
                false, a, false, bm, (short)0, acc[tn], false, false);
        }
    }

    // C/D layout: VGPR j -> row (j + 8*half), col r within the tile
    #pragma unroll
    for (int tn = 0; tn < 4; ++tn) {
        const int   n0 = ngrp * 64 + tn * 16;
        const float bv = bias[n0 + r];
        #pragma unroll
        for (int j = 0; j < 8; ++j) {
            const int row = m0 + j + 8 * half;
            XW[(size_t)row * H_ + n0 + r] = acc[tn][j] + bv;
        }
    }
}

// ---------------------------------------------------------------------------
// Kernel 2: sequential recurrence. 2 blocks, 16 batch rows each.
// ---------------------------------------------------------------------------
#define LDS_STRIDE 258   // 16 rows of h, padded: banks (2r + k) % 64, no conflicts

__global__ __launch_bounds__(256) void rnn_recurrence(const float* __restrict__ U,
                                                      float* __restrict__ Y) {
    __shared__ float h[16 * LDS_STRIDE];

    const int lane = threadIdx.x & 31;
    const int wave = threadIdx.x >> 5;
    const int r    = lane & 15;
    const int half = lane >> 4;
    const int b0   = blockIdx.x * 16;        // batch rows owned by this WG
    const int n0a  = (wave * 2 + 0) * 16;    // this wave's two N-tiles
    const int n0b  = (wave * 2 + 1) * 16;

    // ---- Hoist all B fragments of U into registers (once, reused 2048x) ----
    // ufrag[tn][kk]: lane(half,r) holds U[4*kk+2*half .. +1][n0+r]
    v2f ufrag[2][64];
    #pragma unroll
    for (int kk = 0; kk < 64; ++kk) {
        const int k0 = kk * 4;
        ufrag[0][kk].x = U[(size_t)(k0 + 2 * half)     * H_ + n0a + r];
        ufrag[0][kk].y = U[(size_t)(k0 + 2 * half + 1) * H_ + n0a + r];
        ufrag[1][kk].x = U[(size_t)(k0 + 2 * half)     * H_ + n0b + r];
        ufrag[1][kk].y = U[(size_t)(k0 + 2 * half + 1) * H_ + n0b + r];
    }

    // h_{-1} = 0
    for (int i = threadIdx.x; i < 16 * LDS_STRIDE; i += 256) h[i] = 0.0f;

    // ---- Prefetch XW for t = 0 into registers ----
    float xw[2][8];
    #pragma unroll
    for (int j = 0; j < 8; ++j) {
        const size_t brow = (size_t)(b0 + j + 8 * half) * T_;
        xw[0][j] = Y[(brow + 0) * H_ + n0a + r];
        xw[1][j] = Y[(brow + 0) * H_ + n0b + r];
    }

    __syncthreads();

    for (int t = 0; t < T_; ++t) {
        // consume prefetched XW into the accumulators
        v8f acc[2];
        #pragma unroll
        for (int j = 0; j < 8; ++j) {
            acc[0][j] = xw[0][j];
            acc[1][j] = xw[1][j];
        }

        // issue next step's XW loads now; they overlap the whole K loop.
        // (t+1 locations are untouched at step t; clamp keeps last read legal)
        const int tp = (t + 1 < T_) ? (t + 1) : t;
        #pragma unroll
        for (int j = 0; j < 8; ++j) {
            const size_t brow = (size_t)(b0 + j + 8 * half) * T_;
            xw[0][j] = Y[(brow + tp) * H_ + n0a + r];
            xw[1][j] = Y[(brow + tp) * H_ + n0b + r];
        }

        // acc += h_{t-1} @ U : A from LDS, B register-resident, no global loads
        #pragma unroll
        for (int kk = 0; kk < 64; ++kk) {
            v2f a;
            a.x = h[r * LDS_STRIDE + kk * 4 + 2 * half];
            a.y = h[r * LDS_STRIDE + kk * 4 + 2 * half + 1];
            acc[0] = __builtin_amdgcn_wmma_f32_16x16x4_f32(
                false, a, false, ufrag[0][kk], (short)0, acc[0], false, false);
            acc[1] = __builtin_amdgcn_wmma_f32_16x16x4_f32(
                false, a, false, ufrag[1][kk], (short)0, acc[1], false, false);
        }

        // nonlinearity: 16 independent hardware tanh ops (V_TANH_F32)
        #pragma unroll
        for (int tn = 0; tn < 2; ++tn)
            #pragma unroll
            for (int j = 0; j < 8; ++j)
                acc[tn][j] = hw_tanhf(acc[tn][j]);

        __syncthreads();                      // all reads of h_{t-1} done

        // publish h_t to LDS and to Y[b, t, :]
        #pragma unroll
        for (int tn = 0; tn < 2; ++tn) {
            const int n0 = (wave * 2 + tn) * 16;
            #pragma unroll
            for (int j = 0; j < 8; ++j) {
                const int mrow = j + 8 * half;
                h[mrow * LDS_STRIDE + n0 + r] = acc[tn][j];
                Y[((size_t)(b0 + mrow) * T_ + t) * H_ + n0 + r] = acc[tn][j];
            }
        }
        __syncthreads();                      // h_t visible before next step
    }
}

// ---------------------------------------------------------------------------
extern "C" void kernel_launch(void* const* d_in, const int* in_sizes, int n_in,
                              void* d_out, int out_size, void* d_ws, size_t ws_size,
                              hipStream_t stream) {
    const float* x  = (const float*)d_in[0];  // [B, T, D]
    const float* W  = (const float*)d_in[1];  // [D, H]
    const float* U  = (const float*)d_in[2];  // [H, H]
    const float* bv = (const float*)d_in[3];  // [H]
    float* Y = (float*)d_out;                 // [B, T, H]

    // Phase 1: stage XW + b into Y.
    rnn_xw_gemm<<<2048, 256, 0, stream>>>(x, W, bv, Y);

    // Phase 2: serial scan over T; ordered after phase 1 on the same stream.
    rnn_recurrence<<<2, 256, 0, stream>>>(U, Y);
}